// BatchCosineContrastiveLoss_33397665694264
// MI455X (gfx1250) — compile-verified
//
#include <hip/hip_runtime.h>
#include <hip/hip_bf16.h>
#include <math.h>

// ---------------------------------------------------------------------------
// Fused batch cosine contrastive loss for MI455X (gfx1250, wave32, WMMA).
//   loss = mean_i [ logsumexp_j(cos[i,j]) - cos[i,i] ],
//   cos[i,j] = <o1_i, o2_j> / max(|o1_i||o2_j|, 1e-8),  diag gets -1e-12.
// cos in [-1,1] -> logsumexp needs no max subtraction (sum exp <= N*e).
// Diagonal logits are computed in the norms pass (wave already holds both
// rows), so the WMMA GEMM epilogue is branch-free straight-line math.
// GEMM runs on V_WMMA_F32_16X16X4_F32 (native f32 matrix pipe);
// 32 rows per workgroup: each B fragment feeds 2 independent WMMA
// accumulators (halves L2 B-traffic, pipelines back-to-back WMMAs).
// ---------------------------------------------------------------------------

typedef __attribute__((ext_vector_type(2))) float v2f;
typedef __attribute__((ext_vector_type(8))) float v8f;

#define D_FIXED    256
#define LDS_STRIDE 260   // 256 + 4 pad: float4-store aligned, conflict-free v2f reads
#define TILE_M     16
#define ROW_TILES  2     // 2 x 16 = 32 rows per workgroup
#define WG_ROWS    (TILE_M * ROW_TILES)
#define WG_WAVES   8     // 256 threads -> 8 wave32
#define CHUNK_N    (WG_WAVES * 16)   // 128 columns per chunk iteration

// ------- Kernel 1: per-row norms of o1/o2 AND diagonal cosine logit --------
// One wave32 per row i: computes |o1_i|, |o2_i|, and
// diag[i] = <o1_i,o2_i>/max(|o1_i||o2_i|, 1e-8) - 1e-12.
__global__ __launch_bounds__(256)
void cos_norms_diag_kernel(const float* __restrict__ o1, const float* __restrict__ o2,
                           float* __restrict__ norms, float* __restrict__ diag,
                           int N) {
    const int lane = threadIdx.x & 31;
    const int row  = (blockIdx.x * blockDim.x + threadIdx.x) >> 5;
    if (row >= N) return;
    const float4* __restrict__ v1 = (const float4*)(o1 + (size_t)row * D_FIXED);
    const float4* __restrict__ v2 = (const float4*)(o2 + (size_t)row * D_FIXED);
    float4 a0 = v1[lane], a1 = v1[lane + 32];
    float4 b0 = v2[lane], b1 = v2[lane + 32];
    float s1 = a0.x * a0.x + a0.y * a0.y + a0.z * a0.z + a0.w * a0.w
             + a1.x * a1.x + a1.y * a1.y + a1.z * a1.z + a1.w * a1.w;
    float s2 = b0.x * b0.x + b0.y * b0.y + b0.z * b0.z + b0.w * b0.w
             + b1.x * b1.x + b1.y * b1.y + b1.z * b1.z + b1.w * b1.w;
    float sd = a0.x * b0.x + a0.y * b0.y + a0.z * b0.z + a0.w * b0.w
             + a1.x * b1.x + a1.y * b1.y + a1.z * b1.z + a1.w * b1.w;
#pragma unroll
    for (int off = 16; off > 0; off >>= 1) {
        s1 += __shfl_xor(s1, off, 32);
        s2 += __shfl_xor(s2, off, 32);
        sd += __shfl_xor(sd, off, 32);
    }
    if (lane == 0) {
        const float n1 = sqrtf(s1);
        const float n2 = sqrtf(s2);
        norms[row]     = n1;
        norms[N + row] = n2;
        diag[row] = sd / fmaxf(n1 * n2, 1e-8f) - 1e-12f;
    }
}

// ------- Kernel 2: fused WMMA GEMM + per-row sum(exp(cos)) -----------------
__global__ __launch_bounds__(256)
void cos_loss_rows_kernel(const float* __restrict__ o1, const float* __restrict__ o2,
                          const float* __restrict__ norms,
                          const float* __restrict__ diag,
                          float* __restrict__ row_loss, int N) {
    __shared__ float lAs[WG_ROWS * LDS_STRIDE];
    __shared__ float lSum[WG_ROWS];

    const int tid  = threadIdx.x;
    const int wave = tid >> 5;
    const int lane = tid & 31;
    const int lm   = lane & 15;       // lane within half-wave: column / A-row select
    const int hs   = lane >> 4;       // half select: K sub-pair / C row-half
    const int row0 = blockIdx.x * WG_ROWS;

    // Stage A tile (32 rows x 256 K, f32) into LDS, padded stride.
    {
        const float4* __restrict__ src = (const float4*)(o1 + (size_t)row0 * D_FIXED);
#pragma unroll
        for (int i = tid; i < WG_ROWS * (D_FIXED / 4); i += 256) {
            int r  = i >> 6;          // / 64 float4 per row
            int c4 = i & 63;
            float4 val = src[r * (D_FIXED / 4) + c4];
            *(float4*)&lAs[r * LDS_STRIDE + c4 * 4] = val;
        }
    }
    if (tid < WG_ROWS) lSum[tid] = 0.0f;
    __syncthreads();

    // Per-lane row norms (C layout: VGPR r holds row = t*16 + r + 8*hs, col = lm).
    float rnorm[ROW_TILES][8];
#pragma unroll
    for (int t = 0; t < ROW_TILES; ++t)
#pragma unroll
        for (int r = 0; r < 8; ++r)
            rnorm[t][r] = norms[row0 + t * TILE_M + r + 8 * hs];

    float sumexp[ROW_TILES][8];
#pragma unroll
    for (int t = 0; t < ROW_TILES; ++t)
#pragma unroll
        for (int r = 0; r < 8; ++r) sumexp[t][r] = 0.0f;

    const float* __restrict__ aBase0 = &lAs[lm * LDS_STRIDE + 2 * hs];
    const float* __restrict__ aBase1 = &lAs[(TILE_M + lm) * LDS_STRIDE + 2 * hs];

    for (int n0 = 0; n0 < N; n0 += CHUNK_N) {
        const int ncol = n0 + wave * 16 + lm;   // this lane's output column
        const float* __restrict__ bPtr = o2 + (size_t)ncol * D_FIXED + 2 * hs;

        v8f c0 = {};
        v8f c1 = {};
#pragma unroll 4
        for (int k = 0; k < D_FIXED; k += 4) {
            // B 4x16 frag: lane lm = N, VGPR pair = K {2*hs, 2*hs+1} (B[k][n]=o2[n][k])
            v2f b  = *(const v2f*)&bPtr[k];
            // A 16x4 frags: lane lm = M, VGPR pair = K {2*hs, 2*hs+1}
            v2f a0 = *(const v2f*)&aBase0[k];
            v2f a1 = *(const v2f*)&aBase1[k];
            c0 = __builtin_amdgcn_wmma_f32_16x16x4_f32(
                     false, a0, false, b, (short)0, c0, false, false);
            c1 = __builtin_amdgcn_wmma_f32_16x16x4_f32(
                     false, a1, false, b, (short)0, c1, false, false);
        }

        // Branch-free epilogue: scale to cosine, accumulate exp.
        const float ncnorm = norms[N + ncol];
#pragma unroll
        for (int t = 0; t < ROW_TILES; ++t) {
#pragma unroll
            for (int r = 0; r < 8; ++r) {
                const float den = fmaxf(rnorm[t][r] * ncnorm, 1e-8f);
                // fast reciprocal (v_rcp_f32): denom is O(D) here, ~1 ULP is fine
                const float rcp = __builtin_amdgcn_rcpf(den);
                const float dot = (t == 0) ? c0[r] : c1[r];
                sumexp[t][r] += __expf(dot * rcp);
            }
        }
    }

    // Reduce sum(exp) across the 16 lanes of each half-wave, then LDS atomics
    // across waves/halves (ds_add_f32).
#pragma unroll
    for (int t = 0; t < ROW_TILES; ++t) {
#pragma unroll
        for (int r = 0; r < 8; ++r) {
#pragma unroll
            for (int off = 1; off < 16; off <<= 1)
                sumexp[t][r] += __shfl_xor(sumexp[t][r], off, 32);
        }
    }
    if (lm == 0) {
#pragma unroll
        for (int t = 0; t < ROW_TILES; ++t)
#pragma unroll
            for (int r = 0; r < 8; ++r)
                atomicAdd(&lSum[t * TILE_M + r + 8 * hs], sumexp[t][r]);
    }
    __syncthreads();

    if (tid < WG_ROWS)
        row_loss[row0 + tid] = logf(lSum[tid]) - diag[row0 + tid];
}

// ---------------- Kernel 3: mean over per-row losses -----------------------
__global__ __launch_bounds__(256)
void cos_loss_reduce_kernel(const float* __restrict__ row_loss,
                            float* __restrict__ out, int N) {
    __shared__ float sh[256];
    float s = 0.0f;
    for (int i = threadIdx.x; i < N; i += 256) s += row_loss[i];
    sh[threadIdx.x] = s;
    __syncthreads();
#pragma unroll
    for (int w = 128; w > 0; w >>= 1) {
        if (threadIdx.x < w) sh[threadIdx.x] += sh[threadIdx.x + w];
        __syncthreads();
    }
    if (threadIdx.x == 0) out[0] = sh[0] / (float)N;
}

// ---------------------------------------------------------------------------
extern "C" void kernel_launch(void* const* d_in, const int* in_sizes, int n_in,
                              void* d_out, int out_size, void* d_ws, size_t ws_size,
                              hipStream_t stream) {
    const float* o1 = (const float*)d_in[0];
    const float* o2 = (const float*)d_in[1];
    const int N = in_sizes[0] / D_FIXED;   // 8192

    float* ws       = (float*)d_ws;
    float* norms    = ws;          // [2N]: o1 norms then o2 norms
    float* diag     = ws + 2 * N;  // [N]: diagonal cosine logits
    float* row_loss = ws + 3 * N;  // [N]

    // 1) norms + diagonal logits: one wave32 per row, 8 waves per block
    cos_norms_diag_kernel<<<(N + WG_WAVES - 1) / WG_WAVES, 256, 0, stream>>>(
        o1, o2, norms, diag, N);

    // 2) fused WMMA GEMM + per-row logsumexp pieces: one block per 32 rows
    cos_loss_rows_kernel<<<N / WG_ROWS, 256, 0, stream>>>(
        o1, o2, norms, diag, row_loss, N);

    // 3) mean
    cos_loss_reduce_kernel<<<1, 256, 0, stream>>>(row_loss, (float*)d_out, N);
}